// GraphonAggregator_47828755808715
// MI455X (gfx1250) — compile-verified
//
#include <hip/hip_runtime.h>

// GCN-style aggregation: out[i] = (1/deg_i) * ( x[i] + sum_{e: src=i} w_e * x[dst_e] )
// deg_i = max(1, 1 + sum_{e: src=i} w_e)
//
// Pipeline: bin edges by src into CSR (atomic count + LDS block scan + atomic-cursor
// scatter of packed (dst,w) pairs), then one wave32 per node gathers neighbor rows
// from L2-resident x (indices broadcast wave-wide via ds_bpermute) and writes out
// exactly once. No output atomics.

#define NFEAT 128
#define NF4 (NFEAT / 4) // 32 float4 per row == one wave32, one float4 per lane

// ================================================================ CSR build
// counts has N+1 ints; counts[N] is the global segment allocator.
__global__ __launch_bounds__(256) void k_zero(int* __restrict__ counts, int n1) {
    int i = blockIdx.x * blockDim.x + threadIdx.x;
    if (i < n1) counts[i] = 0;
}

__global__ __launch_bounds__(256) void k_count(const int* __restrict__ esrc,
                                               int* __restrict__ counts, int e) {
    int i = blockIdx.x * blockDim.x + threadIdx.x;
    if (i < e) atomicAdd(&counts[esrc[i]], 1); // global_atomic_add_u32
}

// Per-block LDS scan of 256 counts; one global atomic per block for the base.
// Emits meta[i] = (start, len) packed, plus the scatter cursor.
__global__ __launch_bounds__(256) void k_alloc(const int* __restrict__ counts,
                                               int* __restrict__ counts_total, // &counts[N]
                                               int2* __restrict__ meta,
                                               int* __restrict__ cursor, int n) {
    __shared__ int sh[256];
    __shared__ int sbase;
    int tid = threadIdx.x;
    int i = blockIdx.x * 256 + tid;
    int c = (i < n) ? counts[i] : 0;

    sh[tid] = c;
    __syncthreads();
    for (int off = 1; off < 256; off <<= 1) { // Hillis-Steele inclusive scan
        int t = (tid >= off) ? sh[tid - off] : 0;
        __syncthreads();
        sh[tid] += t;
        __syncthreads();
    }
    int incl = sh[tid];
    if (tid == 255) sbase = atomicAdd(counts_total, incl); // block total = sh[255]
    __syncthreads();
    if (i < n) {
        int start = sbase + incl - c; // exclusive
        meta[i]   = make_int2(start, c);
        cursor[i] = start;
    }
}

// Scatter packed (dst, w) pairs: one b64 store per edge.
__global__ __launch_bounds__(256) void k_scatter_edges(const int* __restrict__ esrc,
                                                       const int* __restrict__ edst,
                                                       const float* __restrict__ ew,
                                                       int* __restrict__ cursor,
                                                       int2* __restrict__ pairs, int e) {
    int i = blockIdx.x * blockDim.x + threadIdx.x;
    if (i < e) {
        int s   = esrc[i];
        int pos = atomicAdd(&cursor[s], 1);
        pairs[pos] = make_int2(edst[i], __float_as_int(ew[i]));
    }
}

// ================================================================ gather (no atomics)
// One wave32 per node: lane L owns float4 L of the 128-float row. Neighbor
// (dst, w) pairs load coalesced (32 at a time, one b64 per lane) and broadcast
// lane->wave via __shfl (ds_bpermute_b32: LDS lane-permute path, no LDS memory).
// 2x unrolled so two independent b128 gathers are in flight per iteration.
__global__ __launch_bounds__(256) void k_gather(const float4* __restrict__ x4,
                                                const int2* __restrict__ meta,
                                                const int2* __restrict__ pairs,
                                                float4* __restrict__ out4, int n) {
    unsigned gid  = blockIdx.x * blockDim.x + threadIdx.x;
    unsigned node = gid >> 5;
    unsigned lane = gid & 31u;
    if (node >= (unsigned)n) return;

    int2 md   = meta[node];  // single broadcast-coalesced b64: (start, len)
    int start = md.x;
    int len   = md.y;

    float4 acc  = x4[(size_t)node * NF4 + lane]; // self-loop, weight 1
    float  wsum = 1.0f;

    for (int base = 0; base < len; base += 32) {
        int m = len - base; if (m > 32) m = 32;
        int2 pr = (int(lane) < m) ? pairs[start + base + lane] : make_int2(0, 0);
        int   d_l = pr.x;                  // lane-resident neighbor indices
        float w_l = __int_as_float(pr.y);  // lane-resident weights (pad lanes: 0)

        // warm the first two rows of this chunk (gfx1250 global_prefetch_b8)
        {
            int d0 = __shfl(d_l, 0, 32);
            __builtin_prefetch((const void*)(x4 + (size_t)d0 * NF4 + lane), 0, 0);
            int d1 = __shfl(d_l, 1, 32);
            __builtin_prefetch((const void*)(x4 + (size_t)d1 * NF4 + lane), 0, 0);
        }

        int j = 0;
        for (; j + 1 < m; j += 2) { // two gathers in flight
            int   d0 = __shfl(d_l, j,     32);
            int   d1 = __shfl(d_l, j + 1, 32);
            float w0 = __shfl(w_l, j,     32);
            float w1 = __shfl(w_l, j + 1, 32);
            float4 v0 = x4[(size_t)d0 * NF4 + lane]; // global_load_b128 (L2 hit)
            float4 v1 = x4[(size_t)d1 * NF4 + lane]; // global_load_b128 (L2 hit)
            wsum  += w0 + w1;
            acc.x += w0 * v0.x; acc.y += w0 * v0.y; acc.z += w0 * v0.z; acc.w += w0 * v0.w;
            acc.x += w1 * v1.x; acc.y += w1 * v1.y; acc.z += w1 * v1.z; acc.w += w1 * v1.w;
        }
        if (j < m) { // tail
            int   d0 = __shfl(d_l, j, 32);
            float w0 = __shfl(w_l, j, 32);
            float4 v0 = x4[(size_t)d0 * NF4 + lane];
            wsum  += w0;
            acc.x += w0 * v0.x; acc.y += w0 * v0.y; acc.z += w0 * v0.z; acc.w += w0 * v0.w;
        }
    }

    float inv = 1.0f / fmaxf(wsum, 1.0f);
    float4 o;
    o.x = acc.x * inv; o.y = acc.y * inv; o.z = acc.z * inv; o.w = acc.w * inv;
    out4[(size_t)node * NF4 + lane] = o; // single clean store per element
}

// ================================================================ fallback (atomic scatter)
__global__ __launch_bounds__(256) void k_init_deg(float* __restrict__ deg, int n) {
    int i = blockIdx.x * blockDim.x + threadIdx.x;
    if (i < n) deg[i] = 1.0f;
}

__global__ __launch_bounds__(256) void k_accum_deg(const int* __restrict__ esrc,
                                                   const float* __restrict__ ew,
                                                   float* __restrict__ deg, int e) {
    int i = blockIdx.x * blockDim.x + threadIdx.x;
    if (i < e) unsafeAtomicAdd(&deg[esrc[i]], ew[i]);
}

__global__ __launch_bounds__(256) void k_selfloop(const float4* __restrict__ x4,
                                                  float* __restrict__ deg,
                                                  float4* __restrict__ out4, int n) {
    unsigned gid  = blockIdx.x * blockDim.x + threadIdx.x;
    unsigned node = gid >> 5;
    unsigned lane = gid & 31u;
    if (node >= (unsigned)n) return;
    float d   = deg[node];
    float inv = 1.0f / fmaxf(d, 1.0f);
    if (lane == 0) deg[node] = inv;
    size_t base = (size_t)node * NF4 + lane;
    float4 v = x4[base];
    float4 o;
    o.x = v.x * inv; o.y = v.y * inv; o.z = v.z * inv; o.w = v.w * inv;
    out4[base] = o;
}

__global__ __launch_bounds__(256) void k_edge_scatter(const int* __restrict__ esrc,
                                                      const int* __restrict__ edst,
                                                      const float* __restrict__ ew,
                                                      const float* __restrict__ invdeg,
                                                      const float4* __restrict__ x4,
                                                      float* __restrict__ out, unsigned e) {
    unsigned long long gid = (unsigned long long)blockIdx.x * blockDim.x + threadIdx.x;
    unsigned edge = (unsigned)(gid >> 5);
    unsigned lane = (unsigned)(gid & 31ull);
    if (edge >= e) return;
    int s = esrc[edge];
    int d = edst[edge];
    float scale = ew[edge] * invdeg[s];
    float4 v = x4[(size_t)d * NF4 + lane];
    float* o = out + (size_t)s * NFEAT + lane * 4;
    unsafeAtomicAdd(o + 0, v.x * scale);
    unsafeAtomicAdd(o + 1, v.y * scale);
    unsafeAtomicAdd(o + 2, v.z * scale);
    unsafeAtomicAdd(o + 3, v.w * scale);
}

// ================================================================ launch
extern "C" void kernel_launch(void* const* d_in, const int* in_sizes, int n_in,
                              void* d_out, int out_size, void* d_ws, size_t ws_size,
                              hipStream_t stream) {
    const float* x  = (const float*)d_in[0];   // (N, 128) f32
    const int*   ei = (const int*)d_in[1];     // (2, E) i32 row-major: [src | dst]
    const float* ew = (const float*)d_in[2];   // (E,)   f32

    const int E = in_sizes[2];
    const int N = in_sizes[0] / NFEAT;

    const int* esrc = ei;
    const int* edst = ei + E;
    float* out = (float*)d_out;
    const int T = 256;

    // Workspace layout for CSR path (16B-aligned chunks):
    //   counts : (N+1) int   (counts[N] = allocator total)
    //   meta   : N int2      (start, len)
    //   cursor : N int
    //   pairs  : E int2      (packed dst, w)
    size_t off = 0;
    auto carve = [&](size_t bytes) { size_t o = off; off += (bytes + 15) & ~(size_t)15; return o; };
    size_t o_counts = carve((size_t)(N + 1) * 4);
    size_t o_meta   = carve((size_t)N * 8);
    size_t o_cursor = carve((size_t)N * 4);
    size_t o_pairs  = carve((size_t)E * 8);

    if (ws_size >= off) {
        char* ws = (char*)d_ws;
        int*  counts = (int*)(ws + o_counts);
        int2* meta   = (int2*)(ws + o_meta);
        int*  cursor = (int*)(ws + o_cursor);
        int2* pairs  = (int2*)(ws + o_pairs);

        k_zero<<<(N + 1 + T - 1) / T, T, 0, stream>>>(counts, N + 1);
        k_count<<<(E + T - 1) / T, T, 0, stream>>>(esrc, counts, E);
        k_alloc<<<(N + T - 1) / T, T, 0, stream>>>(counts, counts + N, meta, cursor, N);
        k_scatter_edges<<<(E + T - 1) / T, T, 0, stream>>>(esrc, edst, ew, cursor, pairs, E);
        {
            unsigned long long threads = (unsigned long long)N * 32ull;
            k_gather<<<(unsigned)((threads + T - 1) / T), T, 0, stream>>>(
                (const float4*)x, meta, pairs, (float4*)out, N);
        }
    } else {
        // Fallback: atomic scatter pipeline (needs only N floats of scratch)
        float* deg = (float*)d_ws;
        k_init_deg<<<(N + T - 1) / T, T, 0, stream>>>(deg, N);
        k_accum_deg<<<(E + T - 1) / T, T, 0, stream>>>(esrc, ew, deg, E);
        {
            unsigned long long threads = (unsigned long long)N * 32ull;
            k_selfloop<<<(unsigned)((threads + T - 1) / T), T, 0, stream>>>(
                (const float4*)x, deg, (float4*)out, N);
        }
        {
            unsigned long long threads = (unsigned long long)E * 32ull;
            k_edge_scatter<<<(unsigned)((threads + T - 1) / T), T, 0, stream>>>(
                esrc, edst, ew, deg, (const float4*)x, out, (unsigned)E);
        }
    }
}